// PoolBlock_47519518163431
// MI455X (gfx1250) — compile-verified
//
#include <hip/hip_runtime.h>

// ---------------- problem constants ----------------
constexpr int N = 100000;
constexpr int E = 300000;
constexpr int C = 128;
constexpr int SORTN = 1 << 19;        // 524288 >= E, power of two for bitonic
constexpr float EPS = 1e-5f;
constexpr float ADD_SCORE = 0.5f;
constexpr int NTILES = N / 16;        // 6250 exactly

typedef float v2f __attribute__((ext_vector_type(2)));
typedef float v8f __attribute__((ext_vector_type(8)));
typedef unsigned long long u64;

// ---------------- workspace layout (bytes) ----------------
constexpr size_t OFF_XN   = 0;                                   // N*C f32
constexpr size_t OFF_SA   = OFF_XN  + (size_t)N * C * 4;         // N f32
constexpr size_t OFF_SB   = OFF_SA  + (size_t)N * 4;             // N f32
constexpr size_t OFF_SUM  = OFF_SB  + (size_t)N * 4;             // C f32
constexpr size_t OFF_SQ   = OFF_SUM + (size_t)C * 4;             // C f32
constexpr size_t OFF_SCL  = OFF_SQ  + (size_t)C * 4;             // C f32
constexpr size_t OFF_SHF  = OFF_SCL + (size_t)C * 4;             // C f32
constexpr size_t OFF_MENC = OFF_SHF + (size_t)C * 4;             // N u32
constexpr size_t OFF_DEN  = OFF_MENC+ (size_t)N * 4;             // N f32
constexpr size_t OFF_ESC  = OFF_DEN + (size_t)N * 4;             // E f32
constexpr size_t OFF_KEYS = (OFF_ESC + (size_t)E * 4 + 255) & ~(size_t)255; // SORTN u64
constexpr size_t OFF_CLU  = OFF_KEYS + (size_t)SORTN * 8;        // N i32
constexpr size_t OFF_MAT  = OFF_CLU + (size_t)N * 4;             // N i32
constexpr size_t OFF_SCO  = OFF_MAT + (size_t)N * 4;             // N f32
constexpr size_t OFF_BS   = OFF_SCO + (size_t)N * 4;             // 512 i32
constexpr size_t OFF_CNT  = OFF_BS  + 512 * 4;                   // few i32

// ---------------- helpers ----------------
__device__ __forceinline__ unsigned encf(float f) {
  unsigned u = __float_as_uint(f);
  return (u & 0x80000000u) ? ~u : (u | 0x80000000u);
}
__device__ __forceinline__ float decf(unsigned e) {
  return __uint_as_float((e & 0x80000000u) ? (e ^ 0x80000000u) : ~e);
}

__global__ void k_fillf(float* p, float v, int n) {
  int i = blockIdx.x * blockDim.x + threadIdx.x;
  if (i < n) p[i] = v;
}
__global__ void k_fillu(unsigned* p, unsigned v, int n) {
  int i = blockIdx.x * blockDim.x + threadIdx.x;
  if (i < n) p[i] = v;
}

// ---------------- 1) column sums for BatchNorm stats ----------------
__global__ void k_colstats(const float* __restrict__ x, float* sumv, float* sumsq) {
  int c = threadIdx.x;               // 128 threads = channels, coalesced rows
  float s = 0.f, q = 0.f;
  for (int r = blockIdx.x; r < N; r += gridDim.x) {
    float v = x[(size_t)r * C + c];
    s += v; q += v * v;
  }
  atomicAdd(&sumv[c], s);
  atomicAdd(&sumsq[c], q);
}

__global__ void k_finalize(const float* gamma, const float* beta,
                           const float* sumv, const float* sumsq,
                           float* scale, float* shift) {
  int c = threadIdx.x;
  float mean = sumv[c] * (1.0f / N);
  float var  = sumsq[c] * (1.0f / N) - mean * mean;
  float sc = gamma[c] * rsqrtf(var + EPS);
  scale[c] = sc;
  shift[c] = beta[c] - mean * sc;
}

// ---------------- 2) normalize (streaming, with global_prefetch) ----------------
__global__ void k_normalize(const float* __restrict__ x, const float* __restrict__ scale,
                            const float* __restrict__ shift, float* __restrict__ xn) {
  int stride = gridDim.x * blockDim.x;
  for (int i = blockIdx.x * blockDim.x + threadIdx.x; i < N * C; i += stride) {
    __builtin_prefetch(&x[i + 8192], 0, 0);   // global_prefetch_b8
    int c = i & (C - 1);
    xn[i] = x[i] * scale[c] + shift[c];
  }
}

// ---------------- 3) per-node score dot products via V_WMMA_F32_16X16X4_F32 ----
// sA[n] = dot(xn[n], W[0:128]); sB[n] = dot(xn[n], W[128:256])
// One wave computes a 16-node tile: D(16x16) = A(16x128) x B(128x16), K in 32
// steps of 4.  A layout (16x4 f32): lane<16 holds (M=lane,K=base+{0,1}),
// lane>=16 holds (M=lane-16,K=base+{2,3}).  B columns 0/1 carry the two weight
// halves; columns 2..15 replicate them (B loaded unconditionally from
// W + (col&1)*C, always in bounds) producing dead values in D columns 2..15
// that are simply never stored -- this keeps the loop branch-free with no EXEC
// manipulation: 2x global_load_b64 + 1x v_wmma per K-step.
__global__ void k_scores(const float* __restrict__ xn, const float* __restrict__ W,
                         float* __restrict__ sA, float* __restrict__ sB) {
  int wid  = threadIdx.x >> 5;
  int lane = threadIdx.x & 31;
  int tile = blockIdx.x * (blockDim.x >> 5) + wid;   // wave-uniform
  if (tile >= NTILES) return;
  int row = lane & 15;      // M for A, N for B/D
  int kh  = lane >> 4;      // which K-pair half
  int col = lane & 15;
  const float* arow = xn + ((size_t)(tile * 16 + row)) * C + 2 * kh;
  const float* brow = W + (col & 1) * C + 2 * kh;    // in-bounds for all lanes
  v8f acc = {};
#pragma unroll
  for (int base = 0; base < C; base += 4) {
    v2f a = *(const v2f*)(arow + base);
    v2f b = *(const v2f*)(brow + base);
    acc = __builtin_amdgcn_wmma_f32_16x16x4_f32(
        /*neg_a=*/false, a, /*neg_b=*/false, b,
        /*c_mod=*/(short)0, acc, /*reuse_a=*/false, /*reuse_b=*/false);
  }
  // D column col, rows kh*8 .. kh*8+7 live in this lane's 8 accumulator VGPRs
  if (col < 2) {
    float* outp = (col == 0) ? sA : sB;
    int nb = tile * 16 + kh * 8;
#pragma unroll
    for (int i = 0; i < 8; ++i) outp[nb + i] = acc[i];
  }
}

// ---------------- 4) edge raw scores + segment max ----------------
__global__ void k_edge_raw(const int* __restrict__ src, const int* __restrict__ dst,
                           const float* __restrict__ sA, const float* __restrict__ sB,
                           const float* __restrict__ bptr,
                           float* __restrict__ esc, unsigned* __restrict__ menc) {
  int j = blockIdx.x * blockDim.x + threadIdx.x;
  if (j >= E) return;
  int d = dst[j];
  float er = sA[src[j]] + sB[d] + bptr[0];
  esc[j] = er;
  atomicMax(&menc[d], encf(er));
}

__global__ void k_edge_exp(const int* __restrict__ dst, const unsigned* __restrict__ menc,
                           float* __restrict__ esc, float* __restrict__ den) {
  int j = blockIdx.x * blockDim.x + threadIdx.x;
  if (j >= E) return;
  int d = dst[j];
  float ee = __expf(esc[j] - decf(menc[d]));
  esc[j] = ee;
  atomicAdd(&den[d], ee);
}

__global__ void k_softmax_keys(const int* __restrict__ dst, const float* __restrict__ den,
                               float* __restrict__ esc, u64* __restrict__ keys) {
  int i = blockIdx.x * blockDim.x + threadIdx.x;
  if (i >= SORTN) return;
  if (i < E) {
    float e = esc[i] / den[dst[i]] + ADD_SCORE;
    esc[i] = e;
    keys[i] = ((u64)(~encf(e)) << 32) | (unsigned)i;  // ascending == descending score
  } else {
    keys[i] = ~0ull;
  }
}

// ---------------- 5) global bitonic sort ----------------
__global__ void k_bitonic(u64* keys, int j, int k) {
  int i = blockIdx.x * blockDim.x + threadIdx.x;
  int ixj = i ^ j;
  if (ixj > i) {
    u64 a = keys[i], b = keys[ixj];
    bool up = ((i & k) == 0);
    if ((a > b) == up) { keys[i] = b; keys[ixj] = a; }
  }
}

// ---------------- 6) sequential greedy matching (bitmap + chunks in LDS) ------
constexpr int MCH = 2048;
__global__ void k_match(const u64* __restrict__ keys,
                        const int* __restrict__ src, const int* __restrict__ dst,
                        const float* __restrict__ esc,
                        int* __restrict__ matchedG, int* __restrict__ cluster,
                        float* __restrict__ score, int* __restrict__ cnt) {
  __shared__ unsigned mbits[(N + 31) / 32];          // 12.5 KB matched bitmap
  __shared__ int  sbuf[MCH], tbuf[MCH];
  __shared__ float ebuf[MCH];
  __shared__ int kSh;
  int t = threadIdx.x;
  for (int i = t; i < (N + 31) / 32; i += blockDim.x) mbits[i] = 0u;
  if (t == 0) kSh = 0;
  __syncthreads();
  int nch = (E + MCH - 1) / MCH;
  for (int ch = 0; ch < nch; ++ch) {
    int base = ch * MCH;
    for (int i = t; i < MCH; i += blockDim.x) {
      int idx = base + i;
      if (idx < E) {
        int j = (int)(keys[idx] & 0xFFFFFFFFu);
        sbuf[i] = src[j];
        tbuf[i] = dst[j];
        ebuf[i] = esc[j];
      }
    }
    __syncthreads();
    if (t == 0) {
      int k = kSh;
      int lim = (E - base < MCH) ? (E - base) : MCH;
      for (int i = 0; i < lim; ++i) {
        int s = sbuf[i], d = tbuf[i];
        unsigned ms = (mbits[s >> 5] >> (s & 31)) & 1u;
        unsigned md = (mbits[d >> 5] >> (d & 31)) & 1u;
        if ((ms | md) == 0u) {
          mbits[s >> 5] |= 1u << (s & 31);
          mbits[d >> 5] |= 1u << (d & 31);
          cluster[s] = k;
          cluster[d] = k;
          score[k] = ebuf[i];
          ++k;
        }
      }
      kSh = k;
    }
    __syncthreads();
  }
  if (t == 0) cnt[0] = kSh;
  __syncthreads();
  for (int n = t; n < N; n += blockDim.x)
    matchedG[n] = (int)((mbits[n >> 5] >> (n & 31)) & 1u);
}

// ---------------- 7) prefix scan of unmatched nodes -> singleton clusters -----
__global__ void k_blocksum(const int* __restrict__ matched, int* __restrict__ bsum) {
  int n = blockIdx.x * 256 + threadIdx.x;
  int um = (n < N && !matched[n]) ? 1 : 0;
  __shared__ int sh[256];
  sh[threadIdx.x] = um;
  __syncthreads();
  for (int off = 128; off > 0; off >>= 1) {
    if (threadIdx.x < off) sh[threadIdx.x] += sh[threadIdx.x + off];
    __syncthreads();
  }
  if (threadIdx.x == 0) bsum[blockIdx.x] = sh[0];
}

__global__ void k_scanblocks(int* bsum, int nb) {
  if (threadIdx.x == 0 && blockIdx.x == 0) {
    int run = 0;
    for (int b = 0; b < nb; ++b) { int v = bsum[b]; bsum[b] = run; run += v; }
  }
}

__global__ void k_assign(const int* __restrict__ matched, int* __restrict__ cluster,
                         const int* __restrict__ bsum, const int* __restrict__ cnt) {
  int n = blockIdx.x * 256 + threadIdx.x;
  int um = (n < N && !matched[n]) ? 1 : 0;
  __shared__ int sh[256];
  sh[threadIdx.x] = um;
  __syncthreads();
  for (int off = 1; off < 256; off <<= 1) {
    int v = (threadIdx.x >= (unsigned)off) ? sh[threadIdx.x - off] : 0;
    __syncthreads();
    sh[threadIdx.x] += v;
    __syncthreads();
  }
  if (um) cluster[n] = cnt[0] + bsum[blockIdx.x] + sh[threadIdx.x] - 1;
}

// ---------------- 8) pooled features ----------------
__global__ void k_pool_scatter(const float* __restrict__ xn, const int* __restrict__ cluster,
                               float* __restrict__ newx) {
  int stride = gridDim.x * blockDim.x;
  for (int i = blockIdx.x * blockDim.x + threadIdx.x; i < N * C; i += stride) {
    int n = i >> 7, c = i & (C - 1);
    atomicAdd(&newx[(size_t)cluster[n] * C + c], xn[i]);
  }
}

__global__ void k_pool_scale(float* __restrict__ newx, const float* __restrict__ score) {
  int stride = gridDim.x * blockDim.x;
  for (int i = blockIdx.x * blockDim.x + threadIdx.x; i < N * C; i += stride)
    newx[i] *= score[i >> 7];
}

// ---------------- 9) edge remap keys + coalesce output ----------------
__global__ void k_cluster_keys(const int* __restrict__ src, const int* __restrict__ dst,
                               const int* __restrict__ cluster, u64* __restrict__ keys) {
  int i = blockIdx.x * blockDim.x + threadIdx.x;
  if (i >= SORTN) return;
  if (i < E) {
    u64 cu = (unsigned)cluster[src[i]];
    u64 cv = (unsigned)cluster[dst[i]];
    keys[i] = (cu << 20) | cv;        // cluster ids < 2^17
  } else {
    keys[i] = ~0ull;
  }
}

__global__ void k_edge_out(const u64* __restrict__ keys, int* __restrict__ eo) {
  int i = blockIdx.x * blockDim.x + threadIdx.x;
  if (i >= E) return;
  u64 key = keys[i];
  int su = (int)(key >> 20);
  int sv = (int)(key & 0xFFFFFu);
  bool first = (i == 0) || (key != keys[i - 1]);
  bool valid = first && (su != sv);
  eo[i]     = valid ? su : -1;
  eo[E + i] = valid ? sv : -1;
}

// ---------------- 10) new_batch scatter (deterministic via atomicMax) ---------
__global__ void k_batch_out(const int* __restrict__ batch, const int* __restrict__ cluster,
                            int* __restrict__ outb) {
  int n = blockIdx.x * blockDim.x + threadIdx.x;
  if (n >= N) return;
  atomicMax(&outb[cluster[n]], batch[n]);
}

// ---------------- host ----------------
static inline int gr(long long n, int b) { return (int)((n + b - 1) / b); }

extern "C" void kernel_launch(void* const* d_in, const int* in_sizes, int n_in,
                              void* d_out, int out_size, void* d_ws, size_t ws_size,
                              hipStream_t stream) {
  (void)in_sizes; (void)n_in; (void)out_size; (void)ws_size;
  const float* x     = (const float*)d_in[0];
  const int*   eidx  = (const int*)d_in[1];
  const int*   batch = (const int*)d_in[2];
  const float* gamma = (const float*)d_in[3];
  const float* beta  = (const float*)d_in[4];
  const float* W     = (const float*)d_in[5];
  const float* bias  = (const float*)d_in[6];
  const int* src = eidx;
  const int* dst = eidx + E;

  char* ws = (char*)d_ws;
  float*    XN   = (float*)(ws + OFF_XN);
  float*    SA   = (float*)(ws + OFF_SA);
  float*    SB   = (float*)(ws + OFF_SB);
  float*    SUM  = (float*)(ws + OFF_SUM);
  float*    SQ   = (float*)(ws + OFF_SQ);
  float*    SCL  = (float*)(ws + OFF_SCL);
  float*    SHF  = (float*)(ws + OFF_SHF);
  unsigned* MENC = (unsigned*)(ws + OFF_MENC);
  float*    DEN  = (float*)(ws + OFF_DEN);
  float*    ESC  = (float*)(ws + OFF_ESC);
  u64*      KEYS = (u64*)(ws + OFF_KEYS);
  int*      CLU  = (int*)(ws + OFF_CLU);
  int*      MAT  = (int*)(ws + OFF_MAT);
  float*    SCO  = (float*)(ws + OFF_SCO);
  int*      BS   = (int*)(ws + OFF_BS);
  int*      CNT  = (int*)(ws + OFF_CNT);

  float* out   = (float*)d_out;
  float* newx  = out;                        // N*C
  int*   oedge = (int*)(out + (size_t)N * C);   // 2*E
  int*   obat  = (int*)(out + (size_t)N * C + 2 * (size_t)E); // N

  // per-call initialization (graph-replayed every call)
  k_fillf<<<gr(2 * C, 256), 256, 0, stream>>>(SUM, 0.f, 2 * C);  // SUM+SQ contiguous
  k_fillu<<<gr(N, 256), 256, 0, stream>>>(MENC, 0u, N);
  k_fillf<<<gr(N, 256), 256, 0, stream>>>(DEN, 0.f, N);
  k_fillf<<<gr(N, 256), 256, 0, stream>>>(SCO, 1.f, N);
  k_fillu<<<1, 64, 0, stream>>>((unsigned*)CNT, 0u, 8);
  k_fillf<<<gr((long long)N * C, 256), 256, 0, stream>>>(newx, 0.f, N * C);
  k_fillu<<<gr(N, 256), 256, 0, stream>>>((unsigned*)obat, 0u, N);

  // BatchNorm
  k_colstats<<<256, C, 0, stream>>>(x, SUM, SQ);
  k_finalize<<<1, C, 0, stream>>>(gamma, beta, SUM, SQ, SCL, SHF);
  k_normalize<<<2048, 256, 0, stream>>>(x, SCL, SHF, XN);

  // per-node score dots on the WMMA pipe
  k_scores<<<gr(NTILES, 8), 256, 0, stream>>>(XN, W, SA, SB);

  // edge softmax by destination
  k_edge_raw<<<gr(E, 256), 256, 0, stream>>>(src, dst, SA, SB, bias, ESC, MENC);
  k_edge_exp<<<gr(E, 256), 256, 0, stream>>>(dst, MENC, ESC, DEN);
  k_softmax_keys<<<gr(SORTN, 256), 256, 0, stream>>>(dst, DEN, ESC, KEYS);

  // sort edges by descending score
  for (int k = 2; k <= SORTN; k <<= 1)
    for (int j = k >> 1; j > 0; j >>= 1)
      k_bitonic<<<SORTN / 256, 256, 0, stream>>>(KEYS, j, k);

  // greedy maximal matching (sequential semantics, bitmap in LDS)
  k_match<<<1, 256, 0, stream>>>(KEYS, src, dst, ESC, MAT, CLU, SCO, CNT);

  // singleton cluster ids via scan of unmatched
  int nb = gr(N, 256);
  k_blocksum<<<nb, 256, 0, stream>>>(MAT, BS);
  k_scanblocks<<<1, 32, 0, stream>>>(BS, nb);
  k_assign<<<nb, 256, 0, stream>>>(MAT, CLU, BS, CNT);

  // pooled features
  k_pool_scatter<<<2048, 256, 0, stream>>>(XN, CLU, newx);
  k_pool_scale<<<2048, 256, 0, stream>>>(newx, SCO);

  // remap + coalesce edges
  k_cluster_keys<<<gr(SORTN, 256), 256, 0, stream>>>(src, dst, CLU, KEYS);
  for (int k = 2; k <= SORTN; k <<= 1)
    for (int j = k >> 1; j > 0; j >>= 1)
      k_bitonic<<<SORTN / 256, 256, 0, stream>>>(KEYS, j, k);
  k_edge_out<<<gr(E, 256), 256, 0, stream>>>(KEYS, oedge);

  // new_batch
  k_batch_out<<<gr(N, 256), 256, 0, stream>>>(batch, CLU, obat);
}